// TableAttention_60842506715467
// MI455X (gfx1250) — compile-verified
//
#include <hip/hip_runtime.h>
#include <hip/hip_bf16.h>

typedef __attribute__((ext_vector_type(2))) float v2f;
typedef __attribute__((ext_vector_type(8))) float v8f;

#define BDIM 1024
#define DDIM 512
#define SDIM 606
#define TDIM 32
#define EXTRA 4
#define PR 26
#define PC 22
#define TR 2
#define TC 15
#define PMAX 576   // EXTRA + PR*PC

// ---------------------------------------------------------------------------
// Kernel 1: qW = query @ W^T   (M=1024, N=512, K=512), fp32 WMMA 16x16x4.
// One wave per 16x16 output tile; 8 waves (256 threads) per block.
// A fragment (16x4 f32): lanes 0-15 hold K=0,1 ; lanes 16-31 hold K=2,3.
// B fragment (4x16 f32): B[k][n] = W[n][k], same contiguous-2-float pattern.
// ---------------------------------------------------------------------------
__global__ __launch_bounds__(256) void qw_gemm_wmma(
    const float* __restrict__ Q,   // [1024, 512]
    const float* __restrict__ W,   // [512, 512]
    float* __restrict__ qW)        // [1024, 512]
{
  const int lane = threadIdx.x & 31;
  const int wave = threadIdx.x >> 5;
  const int tile = blockIdx.x * 8 + wave;       // 2048 tiles total
  const int tileM = tile >> 5;                  // 32 N-tiles per M row
  const int tileN = tile & 31;
  const int mb = tileM * 16;
  const int nb = tileN * 16;
  const int l16 = lane & 15;
  const int hi  = (lane >> 4) & 1;              // 0: K=0,1  1: K=2,3

  const float* qrow = Q + (size_t)(mb + l16) * DDIM + 2 * hi;
  const float* wrow = W + (size_t)(nb + l16) * DDIM + 2 * hi;

  v8f acc0 = {};
  v8f acc1 = {};
#pragma unroll 8
  for (int kb = 0; kb < DDIM; kb += 8) {
    v2f a0 = *(const v2f*)(qrow + kb);
    v2f b0 = *(const v2f*)(wrow + kb);
    acc0 = __builtin_amdgcn_wmma_f32_16x16x4_f32(false, a0, false, b0,
                                                 (short)0, acc0, false, false);
    v2f a1 = *(const v2f*)(qrow + kb + 4);
    v2f b1 = *(const v2f*)(wrow + kb + 4);
    acc1 = __builtin_amdgcn_wmma_f32_16x16x4_f32(false, a1, false, b1,
                                                 (short)0, acc1, false, false);
  }
  v8f acc = acc0 + acc1;

  // C/D layout: VGPR v, lane L -> row (v + 8*(L>=16)), col (L&15)
  float* orow = qW + (size_t)(mb + 8 * hi) * DDIM + nb + l16;
#pragma unroll
  for (int v = 0; v < 8; ++v) orow[(size_t)v * DDIM] = acc[v];
}

// ---------------------------------------------------------------------------
// Kernel 2: per-batch fused  align -> mask -> segmented softmax -> plan
// marginals -> scaled output.  One 256-thread block (8 wave32) per batch.
// This is the bandwidth stage: streams 1.27 GB of memory_bank once.
// ---------------------------------------------------------------------------
__global__ __launch_bounds__(256) void table_attn_fused(
    const float* __restrict__ qW,    // [1024, 512]
    const float* __restrict__ mem,   // [1024, 606, 512]
    const float* __restrict__ plan,  // [1024, 32]
    const int*   __restrict__ mlen,  // [1024]
    const int*   __restrict__ s1t,   // [32, 1024, 1]
    const int*   __restrict__ pri,   // [32, 1024, 26]
    const int*   __restrict__ tri,   // [32, 1024, 2]
    float* __restrict__ out)         // [1024, 606]
{
  const int b    = blockIdx.x;
  const int tid  = threadIdx.x;
  const int lane = tid & 31;
  const int wave = tid >> 5;

  __shared__ float align_s[SDIM + 2];
  __shared__ float pprob[PR];
  __shared__ float tprob[TR];

  // Each lane caches its 16-element slice of qW[b] in registers.
  const float* qwr = qW + (size_t)b * DDIM;
  float4 qw[4];
#pragma unroll
  for (int c = 0; c < 4; ++c)
    qw[c] = *(const float4*)(qwr + c * 128 + lane * 4);

  const int len = mlen[b];
  const float* base = mem + (size_t)b * SDIM * DDIM;

  // align[b, s] = dot(qW[b], mem[b, s]) with length mask
  for (int s = wave; s < SDIM; s += 8) {
    const float* row = base + (size_t)s * DDIM;
    if (s + 8 < SDIM)  // prefetch this wave's next row (global_prefetch_b8)
      __builtin_prefetch(row + 8 * DDIM + lane * 4, 0, 3);
    float acc = 0.0f;
#pragma unroll
    for (int c = 0; c < 4; ++c) {
      float4 h = *(const float4*)(row + c * 128 + lane * 4);
      acc += h.x * qw[c].x + h.y * qw[c].y + h.z * qw[c].z + h.w * qw[c].w;
    }
#pragma unroll
    for (int off = 16; off > 0; off >>= 1)
      acc += __shfl_down(acc, off, 32);
    if (lane == 0)
      align_s[s] = (s < len) ? acc : -__builtin_inff();
  }
  __syncthreads();

  // Segmented softmax in LDS: threads 0..25 -> player rows (22 cols),
  // threads 26..27 -> team rows (15 cols).
  if (tid < PR + TR) {
    const bool is_team = (tid >= PR);
    const int  nc   = is_team ? TC : PC;
    const int  basei = is_team ? (PMAX + (tid - PR) * TC) : (EXTRA + tid * PC);
    float m = -__builtin_inff();
    for (int c = 0; c < nc; ++c) m = fmaxf(m, align_s[basei + c]);
    float ssum = 0.0f;
    for (int c = 0; c < nc; ++c) {
      float e = __expf(align_s[basei + c] - m);
      align_s[basei + c] = e;
      ssum += e;
    }
    float inv = 1.0f / ssum;
    for (int c = 0; c < nc; ++c) align_s[basei + c] *= inv;
  }

  // Plan-weighted row marginals (wave 1: threads 32..59).
  if (tid >= 32 && tid < 32 + PR + TR) {
    const int r = tid - 32;
    const bool team = (r >= PR);
    const int rr = team ? (r - PR) : r;
    float sum = 0.0f;
    for (int t = 0; t < TDIM; ++t) {
      float pa  = plan[(size_t)b * TDIM + t];
      int   tgt = s1t[(size_t)t * BDIM + b];
      int   row_idx = (tgt - EXTRA) / PC;
      bool  isp = (row_idx < PR);
      if (team) {
        if (!isp) sum += pa * (float)tri[((size_t)t * BDIM + b) * TR + rr];
      } else {
        if (isp)  sum += pa * (float)pri[((size_t)t * BDIM + b) * PR + rr];
      }
    }
    if (team) tprob[rr] = sum;
    else      pprob[rr] = sum;
  }
  __syncthreads();

  // Assemble output row: [4 zeros | player_table 572 | team_table 30]
  float* orow = out + (size_t)b * SDIM;
  for (int i = tid; i < SDIM; i += 256) {
    float v;
    if (i < EXTRA)      v = 0.0f;
    else if (i < PMAX)  v = align_s[i] * pprob[(i - EXTRA) / PC];
    else                v = align_s[i] * tprob[(i - PMAX) / TC];
    orow[i] = v;
  }
}

extern "C" void kernel_launch(void* const* d_in, const int* in_sizes, int n_in,
                              void* d_out, int out_size, void* d_ws, size_t ws_size,
                              hipStream_t stream) {
  const float* query = (const float*)d_in[0];   // [1024, 512]
  const float* mem   = (const float*)d_in[1];   // [1024, 606, 512]
  const float* W     = (const float*)d_in[2];   // [512, 512]
  const float* plan  = (const float*)d_in[3];   // [1024, 32]
  const int*   mlen  = (const int*)  d_in[4];   // [1024]
  const int*   s1t   = (const int*)  d_in[5];   // [32, 1024, 1]
  const int*   pri   = (const int*)  d_in[6];   // [32, 1024, 26]
  const int*   tri   = (const int*)  d_in[7];   // [32, 1024, 2]
  float* out = (float*)d_out;                   // [1024, 606]

  float* qW = (float*)d_ws;                     // 1024*512 f32 = 2 MB scratch

  // 2048 tiles (64 x 32), 8 waves per block -> 256 blocks
  qw_gemm_wmma<<<256, 256, 0, stream>>>(query, W, qW);
  table_attn_fused<<<BDIM, 256, 0, stream>>>(qW, mem, plan, mlen, s1t, pri, tri, out);
}